// MRConv2d_6150393168687
// MI455X (gfx1250) — compile-verified
//
#include <hip/hip_runtime.h>
#include <cstdint>

// MRConv2d fused:  rel = max_k(x[:, e0] - x[:, e1]);  out = relu(W @ [x; rel] + b)
// B=4, C=192, N=8192, K=16, C_OUT=192.  Memory-bound => keep exact fp32 WMMA.

#define BBATCH 4
#define CCH    192     // C
#define NNODE  8192    // N
#define KNBR   16      // K neighbors
#define COUT   192
#define KDIM   384     // 2C (GEMM reduction dim)
#define NT     32      // output columns per workgroup
#define CP     388     // padded y-row stride (2C+4): 8B-aligned pairs, conflict-free banks

typedef __attribute__((ext_vector_type(2))) float v2f;
typedef __attribute__((ext_vector_type(8))) float v8f;

// ---------------- Kernel 0a: x [B,C,N] -> xt [B,N,C] (tiled transpose) --------
__global__ __launch_bounds__(256) void transpose_xt(const float* __restrict__ x,
                                                    float* __restrict__ xt) {
  __shared__ float tile[32][33];
  const int b  = blockIdx.z;
  const int c0 = blockIdx.y * 32;
  const int n0 = blockIdx.x * 32;
  const int tx = threadIdx.x, ty = threadIdx.y;
#pragma unroll
  for (int i = 0; i < 4; ++i) {
    const int c = c0 + ty + i * 8;
    tile[ty + i * 8][tx] = x[((size_t)b * CCH + c) * NNODE + n0 + tx];
  }
  __syncthreads();
#pragma unroll
  for (int i = 0; i < 4; ++i) {
    const int n = n0 + ty + i * 8;
    xt[((size_t)b * NNODE + n) * CCH + c0 + tx] = tile[tx][ty + i * 8];
  }
}

// ---------------- Kernel 0b: W [COUT,2C] -> Wp [k/2][m][2] (A-fragment layout) -
__global__ __launch_bounds__(256) void pack_w(const float* __restrict__ W,
                                              float* __restrict__ Wp) {
  const int idx = blockIdx.x * 256 + threadIdx.x;
  if (idx >= KDIM * COUT) return;
  const int k2 = idx / (2 * COUT);
  const int rr = idx % (2 * COUT);
  const int m  = rr >> 1;
  const int rb = rr & 1;
  Wp[idx] = W[(size_t)m * KDIM + 2 * k2 + rb];  // {W[m][2k2], W[m][2k2+1]} pairs
}

// ---------------- Fused gather/max + WMMA GEMM + bias + ReLU ------------------
template <bool XT>
__global__ __launch_bounds__(256) void mrconv_fused(
    const float* __restrict__ xsrc,  // XT ? xt[B,N,C] : x[B,C,N]
    const float* __restrict__ wsrc,  // XT ? Wp        : W[COUT,2C]
    const int*   __restrict__ eidx,  // [2,B,N,K]
    const float* __restrict__ bias,  // [COUT]
    float* __restrict__ out) {       // [B,COUT,N]
  __shared__ float ys[NT * CP];      // y tile: [col][2C] rows, padded
  __shared__ int   e0s[NT * KNBR];
  __shared__ int   e1s[NT * KNBR];

  const int bb   = blockIdx.x / (NNODE / NT);
  const int n0   = (blockIdx.x % (NNODE / NT)) * NT;
  const int tid  = threadIdx.x;
  const int lane = tid & 31;
  const int w    = tid >> 5;

  // ---- load neighbor/center index tiles (fully coalesced) ----
  {
    const int* e0g = eidx + ((size_t)bb * NNODE + n0) * KNBR;
    const int* e1g = eidx + ((size_t)(BBATCH + bb) * NNODE + n0) * KNBR;
    for (int t = tid; t < NT * KNBR; t += 256) { e0s[t] = e0g[t]; e1s[t] = e1g[t]; }
  }
  __syncthreads();

  // ---- Stage 1: y = [xs ; max_k(x_j - x_i)] into LDS ----
  {
    const int colsPerWave = NT / 8;
    for (int cc = 0; cc < colsPerWave; ++cc) {
      const int col = w * colsPerWave + cc;
      const int n   = n0 + col;
      v2f xv[3], mv[3];
#pragma unroll
      for (int i = 0; i < 3; ++i) { mv[i].x = -3.0e38f; mv[i].y = -3.0e38f; }
      if constexpr (XT) {
        const float* xr = xsrc + ((size_t)bb * NNODE + n) * CCH;
#pragma unroll
        for (int i = 0; i < 3; ++i) xv[i] = *(const v2f*)(xr + 2 * lane + 64 * i);
      } else {
#pragma unroll
        for (int i = 0; i < 3; ++i) {
          const int c = 2 * lane + 64 * i;
          xv[i].x = xsrc[((size_t)bb * CCH + c) * NNODE + n];
          xv[i].y = xsrc[((size_t)bb * CCH + c + 1) * NNODE + n];
        }
      }
      for (int k = 0; k < KNBR; ++k) {
        const int j0 = e0s[col * KNBR + k];
        const int j1 = e1s[col * KNBR + k];
        if constexpr (XT) {
          const float* p0 = xsrc + ((size_t)bb * NNODE + j0) * CCH;
          const float* p1 = xsrc + ((size_t)bb * NNODE + j1) * CCH;
          if (k + 1 < KNBR) {  // hide L2 gather latency: global_prefetch_b8
            __builtin_prefetch(xsrc + ((size_t)bb * NNODE + e0s[col * KNBR + k + 1]) * CCH + 2 * lane, 0, 3);
            __builtin_prefetch(xsrc + ((size_t)bb * NNODE + e1s[col * KNBR + k + 1]) * CCH + 2 * lane, 0, 3);
          }
#pragma unroll
          for (int i = 0; i < 3; ++i) {
            const v2f a = *(const v2f*)(p0 + 2 * lane + 64 * i);
            const v2f c2 = *(const v2f*)(p1 + 2 * lane + 64 * i);
            mv[i].x = fmaxf(mv[i].x, a.x - c2.x);
            mv[i].y = fmaxf(mv[i].y, a.y - c2.y);
          }
        } else {
#pragma unroll
          for (int i = 0; i < 3; ++i) {
            const int c = 2 * lane + 64 * i;
            const float a0 = xsrc[((size_t)bb * CCH + c) * NNODE + j0];
            const float a1 = xsrc[((size_t)bb * CCH + c + 1) * NNODE + j0];
            const float b0 = xsrc[((size_t)bb * CCH + c) * NNODE + j1];
            const float b1 = xsrc[((size_t)bb * CCH + c + 1) * NNODE + j1];
            mv[i].x = fmaxf(mv[i].x, a0 - b0);
            mv[i].y = fmaxf(mv[i].y, a1 - b1);
          }
        }
      }
#pragma unroll
      for (int i = 0; i < 3; ++i) {  // conflict-free b64 LDS stores
        const int c = 2 * lane + 64 * i;
        *(v2f*)&ys[col * CP + c]       = xv[i];
        *(v2f*)&ys[col * CP + CCH + c] = mv[i];
      }
    }
  }
  __syncthreads();

  // ---- Stage 2: out[:,ntile] = relu(W @ ytile + b) via V_WMMA_F32_16X16X4_F32
  const int wN    = w & 1;        // 2 column tiles of 16
  const int wM    = w >> 1;       // 4 row groups of 48 (3 mtiles each)
  const int ncol0 = wN * 16;
  const int mbase = wM * 48;
  const int r     = lane & 15;
  const int h     = lane >> 4;

  v8f acc[3];
#pragma unroll
  for (int t = 0; t < 3; ++t) acc[t] = (v8f){0.f, 0.f, 0.f, 0.f, 0.f, 0.f, 0.f, 0.f};

  const float* yrow = &ys[(ncol0 + r) * CP];
#pragma unroll 4
  for (int kk = 0; kk < KDIM; kk += 4) {
    const int krow = kk + 2 * h;                       // B frag: lane=N col, VGPRs K={2h,2h+1}
    const v2f bfrag = *(const v2f*)(yrow + krow);      // ds_load_b64, all 64 banks once
#pragma unroll
    for (int t = 0; t < 3; ++t) {
      const int m = mbase + t * 16 + r;                // A frag: lane=M row, VGPRs K={2h,2h+1}
      v2f afrag;
      if constexpr (XT)
        afrag = *(const v2f*)(wsrc + (size_t)(krow >> 1) * (2 * COUT) + 2 * m);
      else
        afrag = *(const v2f*)(wsrc + (size_t)m * KDIM + krow);
      acc[t] = __builtin_amdgcn_wmma_f32_16x16x4_f32(false, afrag, false, bfrag,
                                                     (short)0, acc[t], false, false);
    }
  }

  // ---- epilogue: bias + ReLU, C/D layout: lanes0-15 M=v, lanes16-31 M=v+8 ----
#pragma unroll
  for (int t = 0; t < 3; ++t) {
#pragma unroll
    for (int v = 0; v < 8; ++v) {
      const int o = mbase + t * 16 + 8 * h + v;
      const int n = n0 + ncol0 + r;
      const float val = acc[t][v] + bias[o];
      out[((size_t)bb * COUT + o) * NNODE + n] = fmaxf(val, 0.0f);
    }
  }
}

extern "C" void kernel_launch(void* const* d_in, const int* in_sizes, int n_in,
                              void* d_out, int out_size, void* d_ws, size_t ws_size,
                              hipStream_t stream) {
  const float* x    = (const float*)d_in[0];
  // d_in[1] = x_0 (unused by forward)
  const int*   eidx = (const int*)d_in[2];
  const float* W    = (const float*)d_in[3];
  const float* bias = (const float*)d_in[4];
  float*       out  = (float*)d_out;

  const size_t xtBytes = (size_t)BBATCH * NNODE * CCH * sizeof(float);
  const size_t wpBytes = (size_t)KDIM * COUT * sizeof(float);

  if (ws_size >= xtBytes + wpBytes) {
    float* xt = (float*)d_ws;
    float* Wp = (float*)((char*)d_ws + xtBytes);
    dim3 tb(32, 8);
    dim3 tg(NNODE / 32, CCH / 32, BBATCH);
    transpose_xt<<<tg, tb, 0, stream>>>(x, xt);
    pack_w<<<(KDIM * COUT + 255) / 256, 256, 0, stream>>>(W, Wp);
    mrconv_fused<true><<<BBATCH * (NNODE / NT), 256, 0, stream>>>(xt, Wp, eidx, bias, out);
  } else {
    mrconv_fused<false><<<BBATCH * (NNODE / NT), 256, 0, stream>>>(x, W, eidx, bias, out);
  }
}